// L2loss_28166395527234
// MI455X (gfx1250) — compile-verified
//
#include <hip/hip_runtime.h>
#include <hip/hip_bf16.h>
#include <math.h>

// Problem constants (fixed by the reference).
#define N_PIX (224 * 224)   // 50176 = 196 * 256
#define NBINS 256

typedef __attribute__((ext_vector_type(2))) float v2f;
typedef __attribute__((ext_vector_type(8))) float v8f;

// ---------------------------------------------------------------------------
// Branchless upper_bound over a 256-entry sorted int array:
// returns #{j : a[j] <= p}, result in [0, 256].
// ---------------------------------------------------------------------------
__device__ __forceinline__ int ub256(const int* __restrict__ a, int p) {
    int pos = 0;
#pragma unroll
    for (int st = 128; st >= 1; st >>= 1) {
        if (a[pos + st - 1] <= p) pos += st;
    }
    // pos in [0,255]; extend to 256 when all elements <= p. Spurious increment
    // impossible: if count<=255 then a[pos]=a[count] > p by definition.
    if (a[pos] <= p) pos++;
    return pos;
}

// ---------------------------------------------------------------------------
// Exact 32-lane wave sum via V_WMMA_F32_16X16X4_F32.
// A[m][0] = v(lane m), A[m][2] = v(lane m+16), A[m][1]=A[m][3]=0; B = ones.
// D[m][n] = v[m] + v[m+16] for every n (B all-ones makes its layout moot).
// Sum of 8 D VGPRs gives rows 0-7 (lanes 0-15) / rows 8-15 (lanes 16-31);
// xor-16 shuffle completes the total in all lanes. All summands are small
// integers -> bit-exact in f32.
// ---------------------------------------------------------------------------
__device__ __forceinline__ float wave_sum_wmma(float v) {
    v2f a; a[0] = v;    a[1] = 0.0f;
    v2f b; b[0] = 1.0f; b[1] = 1.0f;
    v8f c = {0.f, 0.f, 0.f, 0.f, 0.f, 0.f, 0.f, 0.f};
    c = __builtin_amdgcn_wmma_f32_16x16x4_f32(false, a, false, b,
                                              (short)0, c, false, false);
    float t = c[0] + c[1] + c[2] + c[3] + c[4] + c[5] + c[6] + c[7];
    t += __shfl_xor(t, 16, 32);   // wave32 on gfx1250
    return t;
}

// ---------------------------------------------------------------------------
// Kernel A: zero the u64 accumulators; compute the 6 sequential f32 cumsums
// (3 target + 3 output channels), floored to int32, into workspace.
// Sequential order matches the reference cumsum semantics.
// ---------------------------------------------------------------------------
__global__ void hist_prep_kernel(const float* __restrict__ target,
                                 const float* __restrict__ output,
                                 int* __restrict__ cum,
                                 unsigned long long* __restrict__ acc) {
    int tid = threadIdx.x;
    if (tid < 3) acc[tid] = 0ull;
    if (tid < 6) {
        const float* src = (tid < 3) ? (target + tid * NBINS)
                                     : (output + (tid - 3) * NBINS);
        int* dst = cum + tid * NBINS;
        float c = 0.0f;
        for (int j = 0; j < NBINS; ++j) {
            c += src[j];
            dst[j] = (int)floorf(c);   // nonneg -> matches int() truncation
        }
    }
}

// ---------------------------------------------------------------------------
// Kernel B: one position per thread (196 blocks x 256 threads = 50176).
// Stage all 6 cum arrays in LDS, binary-search, apply carryover chains,
// reduce d^2 per channel: exact f32-WMMA wave sums -> LDS -> u64 atomics.
// Integer atomics are order-independent => deterministic across replays.
// ---------------------------------------------------------------------------
__global__ void __launch_bounds__(256)
hist_count_kernel(const int* __restrict__ cum,
                  unsigned long long* __restrict__ acc) {
    __shared__ int s_cum[6][NBINS];
    __shared__ unsigned s_wave[8][3];

    const int tid = threadIdx.x;
    for (int i = tid; i < 6 * NBINS; i += 256)
        ((int*)s_cum)[i] = cum[i];
    __syncthreads();

    const int p = blockIdx.x * 256 + tid;   // exact cover, no tail

    int h1p = 0, h2p = 0;                    // carryover chains (channel 0: zeros)
    int d2[3];
#pragma unroll
    for (int ch = 0; ch < 3; ++ch) {
        int s1 = ub256(s_cum[ch], p);
        int h1 = (s1 == NBINS - 1) ? h1p : min(s1, NBINS - 1);
        int s2 = ub256(s_cum[3 + ch], p);
        int h2 = (s2 == NBINS - 1) ? h2p : min(s2, NBINS - 1);
        int d = h1 - h2;
        d2[ch] = d * d;                      // <= 65025
        h1p = h1; h2p = h2;
    }

    const int wave = tid >> 5;
    const int lane = tid & 31;
#pragma unroll
    for (int ch = 0; ch < 3; ++ch) {
        float t = wave_sum_wmma((float)d2[ch]);   // exact (< 2^24)
        if (lane == 0) s_wave[wave][ch] = (unsigned)t;
    }
    __syncthreads();

    if (tid == 0) {
        unsigned long long a0 = 0, a1 = 0, a2 = 0;
        for (int w = 0; w < 8; ++w) {
            a0 += s_wave[w][0];
            a1 += s_wave[w][1];
            a2 += s_wave[w][2];
        }
        atomicAdd(&acc[0], a0);
        atomicAdd(&acc[1], a1);
        atomicAdd(&acc[2], a2);
    }
}

// ---------------------------------------------------------------------------
// Kernel C: loss = sum of per-channel sqrt of the exact integer SSDs.
// ---------------------------------------------------------------------------
__global__ void hist_finalize_kernel(const unsigned long long* __restrict__ acc,
                                     float* __restrict__ out) {
    if (threadIdx.x == 0 && blockIdx.x == 0) {
        out[0] = sqrtf((float)acc[0]) + sqrtf((float)acc[1]) +
                 sqrtf((float)acc[2]);
    }
}

extern "C" void kernel_launch(void* const* d_in, const int* in_sizes, int n_in,
                              void* d_out, int out_size, void* d_ws, size_t ws_size,
                              hipStream_t stream) {
    const float* target = (const float*)d_in[0];   // (3,256,1) f32
    const float* output = (const float*)d_in[1];   // (3,256,1) f32
    float* out = (float*)d_out;                    // scalar f32

    // Workspace layout: [0,24): 3 x u64 accumulators; [64, 64+6144): 6x256 int cum.
    unsigned long long* acc = (unsigned long long*)d_ws;
    int* cum = (int*)((char*)d_ws + 64);

    hist_prep_kernel<<<1, 64, 0, stream>>>(target, output, cum, acc);
    hist_count_kernel<<<N_PIX / 256, 256, 0, stream>>>(cum, acc);
    hist_finalize_kernel<<<1, 32, 0, stream>>>(acc, out);
}